// IndRNN_onlyrecurrent_58677843198243
// MI455X (gfx1250) — compile-verified
//
#include <hip/hip_runtime.h>

#define LEN     2048
#define BATCH   32
#define HIDDEN  1024
#define NELEM   (BATCH * HIDDEN)       // 32768 independent chains
#define TPB     64                     // 2 waves per block
#define NBLK    (NELEM / TPB)          // 512 blocks
#define STAGES  64                     // timesteps prefetched ahead per wave
#define TILES   (LEN / STAGES)         // 32

// ---- gfx1250 async-copy path detection (device compile only) ----
#if defined(__HIP_DEVICE_COMPILE__) && defined(__gfx1250__)
#if defined(__has_builtin)
#if __has_builtin(__builtin_amdgcn_global_load_async_to_lds_b32)
#define USE_ASYNC 1
#endif
#endif
#endif
#ifndef USE_ASYNC
#define USE_ASYNC 0
#endif

#if USE_ASYNC
// s_wait_asynccnt via inline asm with a compiler memory barrier so the LDS
// consume load cannot be hoisted above the wait. Waiting to <=62 with 64
// potentially outstanding (6-bit counter, in-order completion) proves the
// oldest stage has landed.
#define WAIT_ASYNCCNT(n) asm volatile("s_wait_asynccnt " #n ::: "memory")
typedef __attribute__((address_space(1))) int* as1_ptr;   // builtin arg0: global int*
typedef __attribute__((address_space(3))) int* as3_ptr;   // builtin arg1: shared int*
#endif

__global__ __launch_bounds__(TPB)
void indrnn_recurrent_kernel(const float* __restrict__ x,
                             const float* __restrict__ w_hh,
                             const float* __restrict__ h0,
                             float* __restrict__ out)
{
    const int tid = threadIdx.x;
    const int idx = blockIdx.x * TPB + tid;           // 0..32767, coalesced
    const float w = w_hh[idx & (HIDDEN - 1)];
    float h       = h0[idx];
    const float* xp = x   + idx;                      // stride NELEM per t
    float*       op = out + idx;

#if USE_ASYNC
    // Lane-private LDS ring: stage s, lane tid -> xbuf[s*TPB + tid].
    // Each lane only ever reads the slot it itself async-loaded, so no
    // barriers are required; ASYNCcnt (in-order completion) is the only sync.
    __shared__ float xbuf[STAGES * TPB];              // 16 KB / block

    // Prologue: fill the pipeline (up to 64 outstanding async loads; the
    // 64th issue stalls briefly at the 6-bit counter cap, which is safe).
#pragma unroll
    for (int s = 0; s < STAGES; ++s) {
        __builtin_amdgcn_global_load_async_to_lds_b32(
            (as1_ptr)(xp + (size_t)s * NELEM),
            (as3_ptr)&xbuf[s * TPB + tid],
            0, 0);
    }

    // Steady state: consume stage s of tile k, immediately re-issue the
    // async load for stage s of tile k+1 into the same slot. After the
    // wait (<=62) plus one new issue, outstanding stays <=63: no issue
    // stalls in steady state, ~63 stages (~8 MB GPU-wide) in flight.
    for (int tile = 0; tile < TILES - 1; ++tile) {
        const float* xnext = xp + (size_t)(tile + 1) * STAGES * NELEM;
        float*       otile = op + (size_t)tile * STAGES * NELEM;
#pragma unroll
        for (int s = 0; s < STAGES; ++s) {
            WAIT_ASYNCCNT(62);                        // oldest stage landed
            float xv = xbuf[s * TPB + tid];           // ds_load_b32
            __builtin_amdgcn_global_load_async_to_lds_b32(
                (as1_ptr)(xnext + (size_t)s * NELEM),
                (as3_ptr)&xbuf[s * TPB + tid],
                0, 0);
            h = fmaxf(fmaf(h, w, xv), 0.0f);
            __builtin_nontemporal_store(h, otile + (size_t)s * NELEM);
        }
    }

    // Final tile: everything already in flight; drain once, then consume.
    {
        WAIT_ASYNCCNT(0);
        float* otile = op + (size_t)(TILES - 1) * STAGES * NELEM;
#pragma unroll
        for (int s = 0; s < STAGES; ++s) {
            float xv = xbuf[s * TPB + tid];
            h = fmaxf(fmaf(h, w, xv), 0.0f);
            __builtin_nontemporal_store(h, otile + (size_t)s * NELEM);
        }
    }
#else
    // Fallback: register software pipeline, depth STAGES, loadcnt-tracked.
    float buf[STAGES];
#pragma unroll
    for (int s = 0; s < STAGES; ++s)
        buf[s] = __builtin_nontemporal_load(xp + (size_t)s * NELEM);

    for (int tile = 0; tile < TILES - 1; ++tile) {
        const float* xnext = xp + (size_t)(tile + 1) * STAGES * NELEM;
        float*       otile = op + (size_t)tile * STAGES * NELEM;
#pragma unroll
        for (int s = 0; s < STAGES; ++s) {
            float xv = buf[s];
            buf[s] = __builtin_nontemporal_load(xnext + (size_t)s * NELEM);
            h = fmaxf(fmaf(h, w, xv), 0.0f);
            __builtin_nontemporal_store(h, otile + (size_t)s * NELEM);
        }
    }
    {
        float* otile = op + (size_t)(TILES - 1) * STAGES * NELEM;
#pragma unroll
        for (int s = 0; s < STAGES; ++s) {
            h = fmaxf(fmaf(h, w, buf[s]), 0.0f);
            __builtin_nontemporal_store(h, otile + (size_t)s * NELEM);
        }
    }
#endif
}

extern "C" void kernel_launch(void* const* d_in, const int* in_sizes, int n_in,
                              void* d_out, int out_size, void* d_ws, size_t ws_size,
                              hipStream_t stream)
{
    (void)in_sizes; (void)n_in; (void)out_size; (void)d_ws; (void)ws_size;
    const float* x    = (const float*)d_in[0];   // [LEN, BATCH, HIDDEN]
    const float* whh  = (const float*)d_in[1];   // [HIDDEN]
    const float* h0   = (const float*)d_in[2];   // [BATCH, HIDDEN]
    float*       out  = (float*)d_out;           // [LEN, BATCH, HIDDEN]

    indrnn_recurrent_kernel<<<NBLK, TPB, 0, stream>>>(x, whh, h0, out);
}